// self_attn_83992380441363
// MI455X (gfx1250) — compile-verified
//
#include <hip/hip_runtime.h>
#include <hip/hip_bf16.h>
#include <hip/hip_fp16.h>

// MI455X / gfx1250 fused sigmoid-attention (no softmax), fp32 throughout.
// Matrix math: V_WMMA_F32_16X16X4_F32 (reference-exact fp32 FMA, RNE).
// Data movement: Tensor Data Mover (tensor_load_to_lds) with D# pad fields
// generating bank-conflict-free LDS strides, double-buffered on TENSORcnt.

typedef float v2f __attribute__((ext_vector_type(2)));
typedef float v8f __attribute__((ext_vector_type(8)));
typedef unsigned int u32x4 __attribute__((ext_vector_type(4)));
typedef int i32x4 __attribute__((ext_vector_type(4)));
typedef int i32x8 __attribute__((ext_vector_type(8)));

#define BATCH 4
#define CH    64
#define CQD   8
#define NPOS  4096   // 64*64

// ---------------------------------------------------------------------------
// Kernel 1: QKV 1x1-conv projections.
//   Q[b][n][q]  (q contiguous  -> float2 B-operand loads in kernel 2)
//   K[b][q][n]  (n contiguous  -> TDM tile rows)
//   V[b][o][n]  (n contiguous  -> TDM tile rows)
// ---------------------------------------------------------------------------
__global__ __launch_bounds__(256) void qkv_proj_kernel(
    const float* __restrict__ x,  const float* __restrict__ Wq,
    const float* __restrict__ Wk, const float* __restrict__ Wv,
    float* __restrict__ Q, float* __restrict__ K, float* __restrict__ V)
{
  __shared__ float wq[CQD * CH];
  __shared__ float wk[CQD * CH];
  __shared__ float wv[CH * CH];

  const int tid = threadIdx.x;
  for (int i = tid; i < CQD * CH; i += 256) { wq[i] = Wq[i]; wk[i] = Wk[i]; }
  for (int i = tid; i < CH * CH;  i += 256) { wv[i] = Wv[i]; }
  __syncthreads();

  const int gid = blockIdx.x * 256 + tid;
  const int b   = gid >> 12;        // / NPOS
  const int n   = gid & (NPOS - 1);
  const float* xcol = x + (b * CH) * NPOS + n;

  float aq[CQD], ak[CQD], av[CH];
#pragma unroll
  for (int q = 0; q < CQD; ++q) { aq[q] = 0.f; ak[q] = 0.f; }
#pragma unroll
  for (int o = 0; o < CH; ++o)  { av[o] = 0.f; }

  for (int c = 0; c < CH; ++c) {
    const float xc = xcol[c * NPOS];          // coalesced across lanes
#pragma unroll
    for (int q = 0; q < CQD; ++q) {
      aq[q] = fmaf(wq[q * CH + c], xc, aq[q]);
      ak[q] = fmaf(wk[q * CH + c], xc, ak[q]);
    }
#pragma unroll
    for (int o = 0; o < CH; ++o) {
      av[o] = fmaf(wv[o * CH + c], xc, av[o]);
    }
  }

  float4* qrow = (float4*)(Q + (size_t)(b * NPOS + n) * CQD);
  qrow[0] = make_float4(aq[0], aq[1], aq[2], aq[3]);
  qrow[1] = make_float4(aq[4], aq[5], aq[6], aq[7]);
#pragma unroll
  for (int q = 0; q < CQD; ++q) K[(b * CQD + q) * NPOS + n] = ak[q];
#pragma unroll
  for (int o = 0; o < CH; ++o)  V[(b * CH + o) * NPOS + n]  = av[o];
}

// ---------------------------------------------------------------------------
// TDM 2-D tile load: rows x cols tile of a row-major [tensor_h][4096] f32
// tensor into LDS, with hardware pad after every 16 dwords so LDS row stride
// becomes (32 + 2*pad) dwords (bank-conflict-free layouts without ds math).
//   D# group0: count=1 | lds_addr | global_addr | type=2      (ISA 8.3)
//   D# group1: data_size=4B, pad fields, dims/strides         (ISA 8.4)
// Toolchain ships the 6-arg builtin: (v4u, v8i, v4i, v4i, v8i, i32 cpol);
// groups 2/3 (+extra slot) are zero for a 2-D tensor.
// ---------------------------------------------------------------------------
__device__ __forceinline__ void tdm_load_tile_f32(
    unsigned lds_byte_off, const float* gaddr,
    unsigned tile_w, unsigned tile_h, unsigned tensor_h,
    unsigned pad_amount_code /* pad = code+1 dwords per 16 dwords */)
{
  const unsigned long long ga = (unsigned long long)(uintptr_t)gaddr;

  u32x4 g0;
  g0[0] = 1u;                                        // count=1, no gather
  g0[1] = lds_byte_off;                              // LDS byte address
  g0[2] = (unsigned)(ga & 0xffffffffu);              // global_addr[31:0]
  g0[3] = (unsigned)((ga >> 32) & 0x01ffffffu)       // global_addr[56:32]
        | (2u << 30);                                // type=2 ("image")

  i32x8 g1;
  g1[0] = (int)((2u << 16)                           // data_size = 4 bytes
              | (1u << 20)                           // pad_enable
              | (3u << 22)                           // pad_interval = 16 dw
              | (pad_amount_code << 25));            // pad amount
  g1[1] = (int)((unsigned)NPOS << 16);               // tensor_dim0 lo16
  g1[2] = (int)(tensor_h << 16);                     // dim0 hi | tensor_dim1
  g1[3] = (int)(tile_w << 16);                       // dim1 hi | tile_dim0
  g1[4] = (int)tile_h;                               // tile_dim1 | tile_dim2=0
  g1[5] = (int)NPOS;                                 // tensor_dim0_stride
  g1[6] = 0;
  g1[7] = 0;

  const i32x4 z4 = {};
  const i32x8 z8 = {};
  __builtin_amdgcn_tensor_load_to_lds(g0, g1, z4, z4, z8, 0);
}

// ---------------------------------------------------------------------------
// Kernel 2: fused  out[b,o,m] = gamma * sum_n V[o,n]*sigmoid(q_m . k_n) + x
// 128 threads = 4 waves; wave w owns m-tile [m0,m0+16); block owns 64 m's.
// n-loop steps 32 columns: TDM double-buffers K[8x32] and V[64x32] tiles in
// LDS while all waves run 36 f32-WMMAs per step (4x score, 32x AV-accum).
// ---------------------------------------------------------------------------
#define VROW 36                 // 32 + 2*2 pad dwords   (36r mod 64 distinct)
#define KROW 40                 // 32 + 2*4 pad dwords   (q*40+h*20 disjoint)
#define VBUF (CH * VROW)        // 2304 dwords
#define KBUF (CQD * KROW)       //  320 dwords

__global__ __launch_bounds__(128) void attn_fused_kernel(
    const float* __restrict__ x, const float* __restrict__ Q,
    const float* __restrict__ K, const float* __restrict__ V,
    const float* __restrict__ gamma_p, float* __restrict__ out)
{
  __shared__ float smem[2 * VBUF + 2 * KBUF];        // 21 KB
  // Flat-address low 32 bits of an LDS pointer == LDS byte offset (runtime
  // computation only; never materialized into a static initializer).
  const unsigned lds_base = (unsigned)(uintptr_t)(void*)smem;

  const int tid   = threadIdx.x;
  const int lane  = tid & 31;
  const int wave  = tid >> 5;
  const int b     = blockIdx.x >> 6;    // 64 m-blocks per batch
  const int mblk  = blockIdx.x & 63;
  const int m0    = mblk * 64 + wave * 16;
  const int lhalf = lane >> 4;          // 0: lanes 0-15, 1: lanes 16-31
  const int l16   = lane & 15;

  // B-operand from Q for score WMMAs: VGPR0 = rows q{0,2}, VGPR1 = q{1,3}
  const float* qb = Q + (size_t)(b * NPOS + m0 + l16) * CQD + lhalf * 2;
  const v2f qb01 = *(const v2f*)(qb);       // q = {0,1} / {2,3}
  const v2f qb23 = *(const v2f*)(qb + 4);   // q = {4,5} / {6,7}

  v8f acc[4];                                // O tile: 64 o-rows x 16 m-cols
#pragma unroll
  for (int ot = 0; ot < 4; ++ot) acc[ot] = (v8f){};

  const float* Kb = K + (size_t)b * CQD * NPOS;
  const float* Vb = V + (size_t)b * CH  * NPOS;

  // prologue: wave 0 DMAs the first K/V tiles into buffer 0
  if (wave == 0) {
    tdm_load_tile_f32(lds_base, Vb, 32, CH, CH, 1);                 // V str 36
    tdm_load_tile_f32(lds_base + 2u * VBUF * 4u, Kb, 32, CQD, CQD, 3); // K str 40
  }

  for (int it = 0; it < NPOS / 32; ++it) {
    const int cur = it & 1;
    const int nxt = cur ^ 1;
    if (wave == 0) __builtin_amdgcn_s_wait_tensorcnt(0);
    __syncthreads();   // tiles ready; everyone done reading the other buffer

    if (wave == 0 && it + 1 < NPOS / 32) {   // overlap next DMA with compute
      const int nn = (it + 1) * 32;
      tdm_load_tile_f32(lds_base + (unsigned)nxt * VBUF * 4u,
                        Vb + nn, 32, CH, CH, 1);
      tdm_load_tile_f32(lds_base + (2u * VBUF + (unsigned)nxt * KBUF) * 4u,
                        Kb + nn, 32, CQD, CQD, 3);
    }

    const float* vl = smem + cur * VBUF;
    const float* kl = smem + 2 * VBUF + cur * KBUF;

#pragma unroll
    for (int h = 0; h < 2; ++h) {            // two 16-col score tiles
      // T[16n x 16m] = sigmoid(K^T . Q): A rows = local n, contraction = q
      v2f a01, a23;
      a01.x = kl[(lhalf * 2 + 0) * KROW + h * 20 + l16];
      a01.y = kl[(lhalf * 2 + 1) * KROW + h * 20 + l16];
      a23.x = kl[(lhalf * 2 + 4) * KROW + h * 20 + l16];
      a23.y = kl[(lhalf * 2 + 5) * KROW + h * 20 + l16];

      v8f t = (v8f){};
      t = __builtin_amdgcn_wmma_f32_16x16x4_f32(false, a01, false, qb01,
                                                (short)0, t, false, false);
      t = __builtin_amdgcn_wmma_f32_16x16x4_f32(false, a23, false, qb23,
                                                (short)0, t, false, false);
#pragma unroll
      for (int i = 0; i < 8; ++i)
        t[i] = 1.0f / (1.0f + __expf(-t[i]));

      // O += V[16o x 4n] . T^T[4n x 16m]; T VGPR i = tile-rows {i, i+8};
      // B-operand rows {r,r+2} built with half-wave shfl_xor swaps.
#pragma unroll
      for (int nb = 0; nb < 4; ++nb) {
        const int ti = (nb & 1) * 4;
        float bv0, bv1;
        if (nb < 2) {
          bv0 = (lane < 16) ? t[ti + 0] : __shfl_xor(t[ti + 2], 16, 32);
          bv1 = (lane < 16) ? t[ti + 1] : __shfl_xor(t[ti + 3], 16, 32);
        } else {
          bv0 = (lane < 16) ? __shfl_xor(t[ti + 0], 16, 32) : t[ti + 2];
          bv1 = (lane < 16) ? __shfl_xor(t[ti + 1], 16, 32) : t[ti + 3];
        }
        v2f bb; bb.x = bv0; bb.y = bv1;

#pragma unroll
        for (int ot = 0; ot < 4; ++ot) {
          const v2f av = *(const v2f*)(vl + (ot * 16 + l16) * VROW +
                                       h * 18 + nb * 4 + lhalf * 2);
          acc[ot] = __builtin_amdgcn_wmma_f32_16x16x4_f32(
              false, av, false, bb, (short)0, acc[ot], false, false);
        }
      }
    }
  }

  // ---- epilogue: out = gamma * O + x (straight from C/D VGPR layout) ----
  const float g = gamma_p[0];
#pragma unroll
  for (int ot = 0; ot < 4; ++ot) {
#pragma unroll
    for (int i = 0; i < 8; ++i) {
      const int o = ot * 16 + i + lhalf * 8;
      const int m = m0 + l16;
      const size_t idx = (size_t)(b * CH + o) * NPOS + m;
      out[idx] = fmaf(g, acc[ot][i], x[idx]);
    }
  }
}

// ---------------------------------------------------------------------------
extern "C" void kernel_launch(void* const* d_in, const int* in_sizes, int n_in,
                              void* d_out, int out_size, void* d_ws, size_t ws_size,
                              hipStream_t stream) {
  const float* x     = (const float*)d_in[0];
  const float* Wq    = (const float*)d_in[1];
  const float* Wk    = (const float*)d_in[2];
  const float* Wv    = (const float*)d_in[3];
  const float* gamma = (const float*)d_in[4];
  float* out = (float*)d_out;

  // workspace: Q (B*N*8) + K (B*N*8) + V (B*N*64) floats = 5.25 MB
  float* Q = (float*)d_ws;
  float* K = Q + (size_t)BATCH * NPOS * CQD;
  float* V = K + (size_t)BATCH * NPOS * CQD;

  qkv_proj_kernel<<<BATCH * NPOS / 256, 256, 0, stream>>>(x, Wq, Wk, Wv, Q, K, V);
  attn_fused_kernel<<<BATCH * (NPOS / 64), 128, 0, stream>>>(x, Q, K, V, gamma, out);
}